// Encoder_19885698580667
// MI455X (gfx1250) — compile-verified
//
#include <hip/hip_runtime.h>
#include <math.h>

#define DIM   128
#define LDSP  132   // padded LDS row stride (floats): (4*row + col) % 64 banks -> conflict-free

typedef float v2f __attribute__((ext_vector_type(2)));
typedef float v8f __attribute__((ext_vector_type(8)));

// D = A(16x4,f32) * B(4x16,f32) + C(16x16,f32)   -> v_wmma_f32_16x16x4_f32
__device__ __forceinline__ v8f wmma_f32_k4(v2f a, v2f b, v8f c) {
    return __builtin_amdgcn_wmma_f32_16x16x4_f32(
        /*neg_a=*/false, a, /*neg_b=*/false, b,
        /*c_mod=*/(short)0, c, /*reuse_a=*/false, /*reuse_b=*/false);
}

// ---------------------------------------------------------------------------
// agg := 0  (uint zero == float 0.0f; valid identity since relu outputs >= 0)
// ---------------------------------------------------------------------------
__global__ void __launch_bounds__(256)
zero_u32(unsigned* __restrict__ p, int n) {
    int i = blockIdx.x * 256 + threadIdx.x;
    if (i < n) p[i] = 0u;
}

// ---------------------------------------------------------------------------
// M = relu(H @ Wpool^T + bpool)      [N,128] x [128,128]
// One block: 16 rows; 8 waves each own a 16-column tile; K=128 in steps of 4.
// ---------------------------------------------------------------------------
__global__ void __launch_bounds__(256)
sage_pool_gemm(const float* __restrict__ H, const float* __restrict__ Wpool,
               const float* __restrict__ bpool, float* __restrict__ M, int N) {
    __shared__ float As[16 * LDSP];
    const int tid = threadIdx.x;
    const int rowBase = blockIdx.x * 16;

    #pragma unroll
    for (int i = 0; i < 8; ++i) {
        int t = tid + i * 256;
        int r = t >> 7, c = t & 127;
        int rg = rowBase + r; if (rg >= N) rg = N - 1;
        As[r * LDSP + c] = H[(long)rg * DIM + c];
    }
    __syncthreads();

    const int wave = tid >> 5, lane = tid & 31;
    const int half = lane >> 4, l15 = lane & 15;
    const int colBase = wave * 16;

    v8f acc = {};
    const float* ar = As + l15 * LDSP + 2 * half;
    const float* wr = Wpool + (long)(colBase + l15) * DIM + 2 * half;
    #pragma unroll
    for (int k = 0; k < DIM; k += 4) {
        v2f a = *(const v2f*)(ar + k);
        v2f b = *(const v2f*)(wr + k);
        acc = wmma_f32_k4(a, b, acc);
    }

    const float bj = bpool[colBase + l15];
    #pragma unroll
    for (int v = 0; v < 8; ++v) {
        int m = v + 8 * half;                 // C/D layout: VGPR v -> row v (+8 for hi half)
        int rg = rowBase + m;
        if (rg < N) {
            float y = acc[v] + bj;
            M[(long)rg * DIM + colBase + l15] = fmaxf(y, 0.0f);
        }
    }
}

// ---------------------------------------------------------------------------
// agg[dst] = max(agg[dst], m[src])  per edge, per feature.
// Values are >= 0, so uint-bitpattern atomicMax == float max.
// 128 threads per edge (coalesced feature reads), 2 edges per block.
// ---------------------------------------------------------------------------
__global__ void __launch_bounds__(256)
scatter_max_edges(const float* __restrict__ M, const int* __restrict__ src,
                  const int* __restrict__ dst, unsigned* __restrict__ agg, int E) {
    int e = blockIdx.x * 2 + (threadIdx.x >> 7);
    int c = threadIdx.x & 127;
    if (e < E) {
        int s = src[e];
        int d = dst[e];
        float v = M[(long)s * DIM + c];
        atomicMax(agg + (long)d * DIM + c, __float_as_uint(v));
    }
}

// ---------------------------------------------------------------------------
// out = H @ Wself^T + AGG @ Wneigh^T + b ;  if do_norm: relu then row-L2-norm.
// Block covers 16 complete rows (8 waves x 16 cols) so the norm is block-local.
// ---------------------------------------------------------------------------
__global__ void __launch_bounds__(256)
sage_combine_gemm(const float* __restrict__ H, const float* __restrict__ AGG,
                  const float* __restrict__ Wself, const float* __restrict__ Wneigh,
                  const float* __restrict__ bias, float* __restrict__ out,
                  int N, int do_norm) {
    __shared__ float As[16 * LDSP];
    __shared__ float Ag[16 * LDSP];
    __shared__ float Ys[16 * LDSP];
    __shared__ float red[16][17];
    __shared__ float scl[16];

    const int tid = threadIdx.x;
    const int rowBase = blockIdx.x * 16;

    #pragma unroll
    for (int i = 0; i < 8; ++i) {
        int t = tid + i * 256;
        int r = t >> 7, c = t & 127;
        int rg = rowBase + r; if (rg >= N) rg = N - 1;
        As[r * LDSP + c] = H[(long)rg * DIM + c];
        Ag[r * LDSP + c] = AGG[(long)rg * DIM + c];
    }
    __syncthreads();

    const int wave = tid >> 5, lane = tid & 31;
    const int half = lane >> 4, l15 = lane & 15;
    const int colBase = wave * 16;

    v8f acc = {};
    {   // self term
        const float* ar = As + l15 * LDSP + 2 * half;
        const float* wr = Wself + (long)(colBase + l15) * DIM + 2 * half;
        #pragma unroll
        for (int k = 0; k < DIM; k += 4)
            acc = wmma_f32_k4(*(const v2f*)(ar + k), *(const v2f*)(wr + k), acc);
    }
    {   // neighbor term
        const float* ar = Ag + l15 * LDSP + 2 * half;
        const float* wr = Wneigh + (long)(colBase + l15) * DIM + 2 * half;
        #pragma unroll
        for (int k = 0; k < DIM; k += 4)
            acc = wmma_f32_k4(*(const v2f*)(ar + k), *(const v2f*)(wr + k), acc);
    }

    const float bj = bias[colBase + l15];

    if (!do_norm) {
        #pragma unroll
        for (int v = 0; v < 8; ++v) {
            int m = v + 8 * half;
            int rg = rowBase + m;
            if (rg < N) out[(long)rg * DIM + colBase + l15] = acc[v] + bj;
        }
        return;
    }

    // relu -> stash tile in LDS -> per-row L2 normalize -> write
    #pragma unroll
    for (int v = 0; v < 8; ++v) {
        int m = v + 8 * half;
        Ys[m * LDSP + colBase + l15] = fmaxf(acc[v] + bj, 0.0f);
    }
    __syncthreads();

    const int r = tid >> 4, seg = tid & 15;      // 16 threads per row
    float s = 0.0f;
    #pragma unroll
    for (int j = 0; j < 8; ++j) {
        float y = Ys[r * LDSP + seg * 8 + j];
        s += y * y;
    }
    red[r][seg] = s;
    __syncthreads();

    if (tid < 16) {
        float t2 = 0.0f;
        #pragma unroll
        for (int j = 0; j < 16; ++j) t2 += red[tid][j];
        float nrm = sqrtf(t2);
        scl[tid] = 1.0f / fmaxf(nrm, 1e-12f);
    }
    __syncthreads();

    int rg = rowBase + r;
    if (rg < N) {
        #pragma unroll
        for (int j = 0; j < 8; ++j) {
            int c = seg * 8 + j;
            out[(long)rg * DIM + c] = Ys[r * LDSP + c] * scl[r];
        }
    }
}

// ---------------------------------------------------------------------------
extern "C" void kernel_launch(void* const* d_in, const int* in_sizes, int n_in,
                              void* d_out, int out_size, void* d_ws, size_t ws_size,
                              hipStream_t stream) {
    const float* x      = (const float*)d_in[0];
    const int*   esrc   = (const int*)  d_in[1];
    const int*   edst   = (const int*)  d_in[2];
    const float* Wpool0 = (const float*)d_in[3];
    const float* Wself0 = (const float*)d_in[4];
    const float* Wneigh0= (const float*)d_in[5];
    const float* Wpool1 = (const float*)d_in[6];
    const float* Wself1 = (const float*)d_in[7];
    const float* Wneigh1= (const float*)d_in[8];
    const float* bpool0 = (const float*)d_in[9];
    const float* b0     = (const float*)d_in[10];
    const float* bpool1 = (const float*)d_in[11];
    const float* b1     = (const float*)d_in[12];

    const int N = in_sizes[0] / DIM;
    const int E = in_sizes[1];

    float*    M    = (float*)d_ws;                                       // [N,128]
    unsigned* AGGu = (unsigned*)((char*)d_ws + (size_t)N * DIM * 4);     // [N,128]
    const float* AGGf = (const float*)AGGu;

    float* out_h   = (float*)d_out;               // return[0]: final conv output
    float* out_enc = out_h + (size_t)N * DIM;     // return[1]: enc_feat_input

    const int rowTiles = (N + 15) / 16;           // 3125 for N=50000
    const int zn = N * DIM;
    dim3 blk(256);

    // ---------------- layer 0: conv -> relu -> l2norm ----------------
    zero_u32<<<(zn + 255) / 256, blk, 0, stream>>>(AGGu, zn);
    sage_pool_gemm<<<rowTiles, blk, 0, stream>>>(x, Wpool0, bpool0, M, N);
    scatter_max_edges<<<(E + 1) / 2, blk, 0, stream>>>(M, esrc, edst, AGGu, E);
    sage_combine_gemm<<<rowTiles, blk, 0, stream>>>(x, AGGf, Wself0, Wneigh0, b0,
                                                    out_enc, N, /*do_norm=*/1);

    // ---------------- layer 1: conv only ----------------
    zero_u32<<<(zn + 255) / 256, blk, 0, stream>>>(AGGu, zn);
    sage_pool_gemm<<<rowTiles, blk, 0, stream>>>(out_enc, Wpool1, bpool1, M, N);
    scatter_max_edges<<<(E + 1) / 2, blk, 0, stream>>>(M, esrc, edst, AGGu, E);
    sage_combine_gemm<<<rowTiles, blk, 0, stream>>>(out_enc, AGGf, Wself1, Wneigh1, b1,
                                                    out_h, N, /*do_norm=*/0);
}